// CPUForgetMult_19000935318056
// MI455X (gfx1250) — compile-verified
//
#include <hip/hip_runtime.h>
#include <cstdint>

// h_t = f_t*x_t + (1-f_t)*h_{t-1}  over T=1024, B*H=65536 independent channels.
// Memory-bound streaming scan: async global->LDS pipeline (ASYNCcnt), depth 31
// (62 outstanding async ops/wave, max legal under the 6-bit ASYNCcnt).

#define T_STEPS 1024
#define BH      65536                 // B*H
#define TPB     256                   // 8 wave32s per block
#define DEPTH   31                    // pipeline depth (2*DEPTH = 62 <= 63 ASYNCcnt)
#define SLOT_FLOATS (TPB * 2)         // f,x interleaved per thread
#define SLOT_BYTES  (SLOT_FLOATS * 4) // 2 KB per timestep slot
#define WAIT_MAIN   (2 * (DEPTH - 1)) // 60

#define STR_(x) #x
#define STR(x) STR_(x)

__global__ __launch_bounds__(TPB)
void forgetmult_scan_kernel(const float* __restrict__ f,
                            const float* __restrict__ x,
                            const float* __restrict__ h0,
                            float* __restrict__ out) {
  // DEPTH timestep slots; each slot: [thread][ {f, x} ] -> 8B per thread
  __shared__ float s_fx[DEPTH * SLOT_FLOATS];

  const int chan = blockIdx.x * TPB + threadIdx.x;     // 0 .. BH-1 (exact)
  const uint64_t stride_b = (uint64_t)BH * sizeof(float);

  const uint64_t fa = (uint64_t)(uintptr_t)(f + chan);
  const uint64_t xa = (uint64_t)(uintptr_t)(x + chan);
  // low 32 bits of a generic pointer to LDS == LDS byte offset
  const uint32_t lds_me =
      (uint32_t)(uintptr_t)(&s_fx[(size_t)threadIdx.x * 2]);

  // ---- prologue: fill all DEPTH slots with async global->LDS copies ----
  // NT: f/x are 512 MB streamed exactly once (> 192 MB L2) -> evict-first.
  #pragma unroll
  for (int t = 0; t < DEPTH; ++t) {
    const uint32_t lf = lds_me + (uint32_t)t * SLOT_BYTES;
    asm volatile("global_load_async_to_lds_b32 %0, %1, off th:TH_LOAD_NT"
                 :: "v"(lf),      "v"(fa + (uint64_t)t * stride_b) : "memory");
    asm volatile("global_load_async_to_lds_b32 %0, %1, off th:TH_LOAD_NT"
                 :: "v"(lf + 4u), "v"(xa + (uint64_t)t * stride_b) : "memory");
  }

  float h = h0[chan];
  int slot = 0;

  // ---- main loop: wait oldest slot, consume, refill, compute, store ----
  for (int t = 0; t < T_STEPS - DEPTH; ++t) {
    // oldest slot's two loads have landed when ASYNCcnt <= 2*(DEPTH-1)
    asm volatile("s_wait_asynccnt " STR(WAIT_MAIN) ::: "memory");
    const float2 v = *(const float2*)(
        &s_fx[(size_t)slot * SLOT_FLOATS + (size_t)threadIdx.x * 2]);
    // ensure the ds read drained before the refill re-targets this slot
    asm volatile("s_wait_dscnt 0" ::: "memory");

    const int tn = t + DEPTH;                      // always < T_STEPS here
    const uint32_t lf = lds_me + (uint32_t)slot * SLOT_BYTES;
    asm volatile("global_load_async_to_lds_b32 %0, %1, off th:TH_LOAD_NT"
                 :: "v"(lf),      "v"(fa + (uint64_t)tn * stride_b) : "memory");
    asm volatile("global_load_async_to_lds_b32 %0, %1, off th:TH_LOAD_NT"
                 :: "v"(lf + 4u), "v"(xa + (uint64_t)tn * stride_b) : "memory");

    // h = f*x + (1-f)*h  ==  h + f*(x - h)   (one sub + one fma)
    h = fmaf(v.x, v.y - h, h);
    __builtin_nontemporal_store(h, out + (size_t)t * BH + chan);

    slot = (slot + 1 == DEPTH) ? 0 : slot + 1;
  }

  // ---- epilogue: all remaining slots already in flight; drain them ----
  asm volatile("s_wait_asynccnt 0" ::: "memory");
  for (int t = T_STEPS - DEPTH; t < T_STEPS; ++t) {
    const float2 v = *(const float2*)(
        &s_fx[(size_t)slot * SLOT_FLOATS + (size_t)threadIdx.x * 2]);
    h = fmaf(v.x, v.y - h, h);
    __builtin_nontemporal_store(h, out + (size_t)t * BH + chan);
    slot = (slot + 1 == DEPTH) ? 0 : slot + 1;
  }
}

extern "C" void kernel_launch(void* const* d_in, const int* in_sizes, int n_in,
                              void* d_out, int out_size, void* d_ws, size_t ws_size,
                              hipStream_t stream) {
  const float* f  = (const float*)d_in[0];   // [T, B, H]
  const float* x  = (const float*)d_in[1];   // [T, B, H]
  const float* h0 = (const float*)d_in[2];   // [B, H]
  float* out = (float*)d_out;                // [T, B, H]

  dim3 grid(BH / TPB);   // 256 blocks
  dim3 block(TPB);       // 256 threads = 8 wave32s
  hipLaunchKernelGGL(forgetmult_scan_kernel, grid, block, 0, stream,
                     f, x, h0, out);
}